// unitary_projection_9637906612749
// MI455X (gfx1250) — compile-verified
//
#include <hip/hip_runtime.h>
#include <hip/hip_bf16.h>
#include <math.h>

// ---------------------------------------------------------------------------
// Problem: N=16384, IN=64, C=4, H=16.
//   Askew = (A - A^H)/2  (complex, per (k,c) 16x16)
//   AX[n,c] = sum_k dX[n,k] * Askew[k,c]          (16x16 complex)
//   s = ceil(log2(||AX||_1)) if >1 else 0 ; P = expm(AX * 2^-s) ; square s times
// Output: (N,C,16,16) complex64 -> interleaved (re,im) float pairs in d_out.
// ---------------------------------------------------------------------------

typedef __attribute__((ext_vector_type(2))) float v2f;
typedef __attribute__((ext_vector_type(8))) float v8f;

#define NMAT   65536       // N*C
#define COLS   1024        // C*H*H
#define KDIM   64          // IN
#define NTAY   12          // Taylor terms (||M||<=1 -> err << f32 eps)

__device__ inline v8f wmma4_f32(v2f a, v2f b, v8f c) {
  // V_WMMA_F32_16X16X4_F32 : D = A(16x4) * B(4x16) + C(16x16)
  return __builtin_amdgcn_wmma_f32_16x16x4_f32(
      /*neg_a=*/false, a, /*neg_b=*/false, b,
      /*c_mod=*/(short)0, c, /*reuse_a=*/false, /*reuse_b=*/false);
}

__device__ inline void lds_fence() {
  // per-wave LDS ordering point (DS ops are in-order within a wave;
  // this blocks compiler reordering and waits outstanding DS ops)
  asm volatile("s_wait_dscnt 0" ::: "memory");
}

// ---------------------------------------------------------------------------
// Kernel 0: skew-Hermitian projection into planar workspace
//   Bre[k*1024 + c*256 + i*16 + j] = 0.5*(Ar[k,c,i,j] - Ar[k,c,j,i])
//   Bim[...]                       = 0.5*(Ai[k,c,i,j] + Ai[k,c,j,i])
// ---------------------------------------------------------------------------
__global__ __launch_bounds__(256) void skew_kernel(
    const float* __restrict__ Ar, const float* __restrict__ Ai,
    float* __restrict__ Bre, float* __restrict__ Bim) {
  int idx = blockIdx.x * 256 + threadIdx.x;      // 65536 elements
  int ij  = idx & 255;
  int i   = ij >> 4, j = ij & 15;
  int tr  = (idx & ~255) + j * 16 + i;           // transposed (i<->j) index
  Bre[idx] = 0.5f * (Ar[idx] - Ar[tr]);
  Bim[idx] = 0.5f * (Ai[idx] + Ai[tr]);
}

// ---------------------------------------------------------------------------
// Kernel 1: AX = dX @ Bsk  (two real GEMMs 16384x64 @ 64x1024, WMMA f32)
// One wave per 16x16 output tile (1024 n-tiles x 64 col-tiles), K=64 -> 16
// chained wmma per component. Writes interleaved complex into d_out.
// ---------------------------------------------------------------------------
__global__ __launch_bounds__(128) void ax_gemm_kernel(
    const float* __restrict__ dX, const float* __restrict__ Bre,
    const float* __restrict__ Bim, float* __restrict__ out) {
  const int lane = threadIdx.x & 31;
  const int w    = threadIdx.x >> 5;
  const int tile = blockIdx.x * 4 + w;           // 65536 tiles
  const int nt   = tile >> 6;                    // n-tile   [0,1024)
  const int ct   = tile & 63;                    // col-tile [0,64)
  const int half = lane >> 4;                    // 0: lanes 0-15, 1: lanes 16-31
  const int l15  = lane & 15;
  const int koff = half << 1;                    // A/B layout: K pair offset
  const int ccol = ct * 16 + l15;

  v8f accR = {}; v8f accI = {};
  for (int q = 0; q < 16; ++q) {                 // K chunks of 4
    int k = 4 * q + koff;
    // A layout (16x4 f32): lane<16 holds row=lane, K=k..k+1; lane>=16 row=lane-16, K=k+2..k+3
    v2f a = *(const v2f*)(dX + (nt * 16 + l15) * KDIM + k);
    v2f br, bi;
    br[0] = Bre[(k    ) * COLS + ccol];
    br[1] = Bre[(k + 1) * COLS + ccol];
    bi[0] = Bim[(k    ) * COLS + ccol];
    bi[1] = Bim[(k + 1) * COLS + ccol];
    accR = wmma4_f32(a, br, accR);
    accI = wmma4_f32(a, bi, accI);
  }
  // C/D layout: lane<16 -> rows 0-7 col=lane ; lane>=16 -> rows 8-15 col=lane-16
  const int r0 = half * 8;
  for (int r = 0; r < 8; ++r) {
    int n = nt * 16 + r0 + r;
    ((float2*)out)[(size_t)n * COLS + ccol] = make_float2(accR[r], accI[r]);
  }
}

// ---------------------------------------------------------------------------
// Kernel 2: per-(n,c) scaling + Taylor expm + repeated squaring.
// One wave per matrix. Complex 16x16 matmul = 4 real WMMA chains of 4.
// LDS (17-float padded rows) used to convert between WMMA register layouts.
// ---------------------------------------------------------------------------
__global__ __launch_bounds__(128) void expm_kernel(float* __restrict__ out) {
  const int lane = threadIdx.x & 31;
  const int w    = threadIdx.x >> 5;
  const int gid  = blockIdx.x * 4 + w;           // (n*4 + c) in [0, 65536)
  const int half = lane >> 4;
  const int l15  = lane & 15;
  const int koff = half << 1;

  __shared__ float sm[4][4][16 * 17];            // [wave][Mre,Mim,Pre,Pim][row*17+col]
  float* Mre = sm[w][0];
  float* Mim = sm[w][1];
  float* Pre = sm[w][2];
  float* Pim = sm[w][3];

  float2* base = ((float2*)out) + (size_t)gid * 256;

  // --- load AX into LDS (coalesced float2 loads) ---
  float2 v[8];
  #pragma unroll
  for (int t = 0; t < 8; ++t) v[t] = base[t * 32 + lane];
  #pragma unroll
  for (int t = 0; t < 8; ++t) {
    int e = t * 32 + lane, r = e >> 4, c = e & 15;
    Mre[r * 17 + c] = v[t].x;
    Mim[r * 17 + c] = v[t].y;
  }
  lds_fence();

  // --- induced 1-norm: max over columns of sum_i |AX[i,j]| ---
  float cs = 0.0f;
  #pragma unroll
  for (int i = 0; i < 16; ++i) {
    float re = Mre[i * 17 + l15], im = Mim[i * 17 + l15];
    cs += sqrtf(re * re + im * im);
  }
  #pragma unroll
  for (int off = 1; off < 16; off <<= 1)
    cs = fmaxf(cs, __shfl_xor(cs, off, 32));
  int s = (cs > 1.0f) ? (int)ceilf(log2f(cs)) : 0;
  float scale = exp2f((float)(-s));

  // --- rescale M in LDS ---
  #pragma unroll
  for (int t = 0; t < 8; ++t) {
    int e = t * 32 + lane, r = e >> 4, c = e & 15;
    Mre[r * 17 + c] *= scale;
    Mim[r * 17 + c] *= scale;
  }
  lds_fence();

  // --- M in A layout (fixed across Taylor loop), plus negated Im copy ---
  v2f aRe[4], aIm[4], nIm[4];
  #pragma unroll
  for (int q = 0; q < 4; ++q) {
    int k = 4 * q + koff;
    aRe[q][0] = Mre[l15 * 17 + k];     aRe[q][1] = Mre[l15 * 17 + k + 1];
    aIm[q][0] = Mim[l15 * 17 + k];     aIm[q][1] = Mim[l15 * 17 + k + 1];
    nIm[q][0] = -aIm[q][0];            nIm[q][1] = -aIm[q][1];
  }

  // --- P = I in LDS ---
  #pragma unroll
  for (int t = 0; t < 8; ++t) {
    int e = t * 32 + lane, r = e >> 4, c = e & 15;
    Pre[r * 17 + c] = (r == c) ? 1.0f : 0.0f;
    Pim[r * 17 + c] = 0.0f;
  }
  lds_fence();

  const int r0 = half * 8;
  float pr[8], pi[8];

  // --- Horner Taylor: P <- I + (M @ P) / k, k = NTAY..1 ---
  for (int k = NTAY; k >= 1; --k) {
    v2f bRe[4], bIm[4];
    #pragma unroll
    for (int q = 0; q < 4; ++q) {      // B layout (4x16): lane=col holds K pair
      int kr = 4 * q + koff;
      bRe[q][0] = Pre[(kr    ) * 17 + l15];
      bRe[q][1] = Pre[(kr + 1) * 17 + l15];
      bIm[q][0] = Pim[(kr    ) * 17 + l15];
      bIm[q][1] = Pim[(kr + 1) * 17 + l15];
    }
    v8f accR = {}; v8f accI = {};
    #pragma unroll
    for (int q = 0; q < 4; ++q) accR = wmma4_f32(aRe[q], bRe[q], accR);
    #pragma unroll
    for (int q = 0; q < 4; ++q) accR = wmma4_f32(nIm[q], bIm[q], accR);
    #pragma unroll
    for (int q = 0; q < 4; ++q) accI = wmma4_f32(aRe[q], bIm[q], accI);
    #pragma unroll
    for (int q = 0; q < 4; ++q) accI = wmma4_f32(aIm[q], bRe[q], accI);
    float invk = 1.0f / (float)k;
    lds_fence();
    #pragma unroll
    for (int r = 0; r < 8; ++r) {
      int row = r0 + r;
      pr[r] = accR[r] * invk + ((row == l15) ? 1.0f : 0.0f);
      pi[r] = accI[r] * invk;
      Pre[row * 17 + l15] = pr[r];
      Pim[row * 17 + l15] = pi[r];
    }
    lds_fence();
  }

  // --- repeated squaring: P <- P @ P, min(s,16) times (wave-uniform) ---
  int nsq = (s > 16) ? 16 : s;
  for (int it = 0; it < nsq; ++it) {
    #pragma unroll
    for (int q = 0; q < 4; ++q) {      // reload P in A layout
      int k = 4 * q + koff;
      aRe[q][0] = Pre[l15 * 17 + k];   aRe[q][1] = Pre[l15 * 17 + k + 1];
      aIm[q][0] = Pim[l15 * 17 + k];   aIm[q][1] = Pim[l15 * 17 + k + 1];
      nIm[q][0] = -aIm[q][0];          nIm[q][1] = -aIm[q][1];
    }
    v2f bRe[4], bIm[4];
    #pragma unroll
    for (int q = 0; q < 4; ++q) {      // P in B layout
      int kr = 4 * q + koff;
      bRe[q][0] = Pre[(kr    ) * 17 + l15];
      bRe[q][1] = Pre[(kr + 1) * 17 + l15];
      bIm[q][0] = Pim[(kr    ) * 17 + l15];
      bIm[q][1] = Pim[(kr + 1) * 17 + l15];
    }
    v8f accR = {}; v8f accI = {};
    #pragma unroll
    for (int q = 0; q < 4; ++q) accR = wmma4_f32(aRe[q], bRe[q], accR);
    #pragma unroll
    for (int q = 0; q < 4; ++q) accR = wmma4_f32(nIm[q], bIm[q], accR);
    #pragma unroll
    for (int q = 0; q < 4; ++q) accI = wmma4_f32(aRe[q], bIm[q], accI);
    #pragma unroll
    for (int q = 0; q < 4; ++q) accI = wmma4_f32(aIm[q], bRe[q], accI);
    lds_fence();
    #pragma unroll
    for (int r = 0; r < 8; ++r) {
      int row = r0 + r;
      pr[r] = accR[r];
      pi[r] = accI[r];
      Pre[row * 17 + l15] = pr[r];
      Pim[row * 17 + l15] = pi[r];
    }
    lds_fence();
  }

  // --- write P interleaved complex ---
  #pragma unroll
  for (int r = 0; r < 8; ++r) {
    int row = r0 + r;
    base[row * 16 + l15] = make_float2(pr[r], pi[r]);
  }
}

// ---------------------------------------------------------------------------
extern "C" void kernel_launch(void* const* d_in, const int* in_sizes, int n_in,
                              void* d_out, int out_size, void* d_ws, size_t ws_size,
                              hipStream_t stream) {
  (void)in_sizes; (void)n_in; (void)out_size; (void)ws_size;
  const float* dX = (const float*)d_in[0];   // (16384, 64)
  const float* Ar = (const float*)d_in[1];   // (64, 4, 16, 16)
  const float* Ai = (const float*)d_in[2];   // (64, 4, 16, 16)
  float* out = (float*)d_out;                // (16384,4,16,16) complex -> interleaved

  float* Bre = (float*)d_ws;                 // 64*1024 floats (256 KB)
  float* Bim = Bre + KDIM * COLS;            // 64*1024 floats (256 KB)

  skew_kernel<<<256, 256, 0, stream>>>(Ar, Ai, Bre, Bim);
  ax_gemm_kernel<<<NMAT / 4, 128, 0, stream>>>(dX, Bre, Bim, out);
  expm_kernel<<<NMAT / 4, 128, 0, stream>>>(out);
}